// FPS_kNN_pytorch3d_38654705664693
// MI455X (gfx1250) — compile-verified
//
#include <hip/hip_runtime.h>

typedef __attribute__((ext_vector_type(2))) float v2f;
typedef __attribute__((ext_vector_type(8))) float v8f;

constexpr int Bn = 4;      // batches
constexpr int Nn = 16384;  // points
constexpr int Gn = 1024;   // FPS groups
constexpr int Kn = 32;     // neighbors

// ---------------------------------------------------------------------------
// 64-bit wave32 xor-shuffle (two 32-bit halves)
// ---------------------------------------------------------------------------
__device__ __forceinline__ unsigned long long shfl_xor_u64(unsigned long long v, int mask) {
    unsigned int lo = (unsigned int)(v & 0xffffffffull);
    unsigned int hi = (unsigned int)(v >> 32);
    lo = __shfl_xor(lo, mask, 32);
    hi = __shfl_xor(hi, mask, 32);
    return ((unsigned long long)hi << 32) | (unsigned long long)lo;
}

// ---------------------------------------------------------------------------
// Kernel 1: farthest point sampling. One 1024-thread block per batch.
// Each thread owns 16 points (strided) held entirely in VGPRs.
// argmax key = (float_bits(dist) << 32) | ~index  -> max picks largest dist,
// ties resolved toward the smallest index (matches jnp.argmax).
// ---------------------------------------------------------------------------
__global__ __launch_bounds__(1024) void fps_kernel(const float* __restrict__ xyz,
                                                   int* __restrict__ fps_idx) {
    const int b   = blockIdx.x;
    const int tid = threadIdx.x;                 // 0..1023
    const float* pts = xyz + (size_t)b * Nn * 3;
    int* out = fps_idx + b * Gn;

    float px[16], py[16], pz[16], mind[16];

    const float sx = pts[0], sy = pts[1], sz = pts[2];

#pragma unroll
    for (int k = 0; k < 16; ++k) {
        const int p = tid + (k << 10);
        const float x = pts[p * 3 + 0];
        const float y = pts[p * 3 + 1];
        const float z = pts[p * 3 + 2];
        px[k] = x; py[k] = y; pz[k] = z;
        const float dx = x - sx, dy = y - sy, dz = z - sz;
        mind[k] = dx * dx + dy * dy + dz * dz;
    }
    if (tid == 0) out[0] = 0;

    __shared__ unsigned long long red[32];
    __shared__ int winner;

    for (int it = 1; it < Gn; ++it) {
        // local argmax over the 16 register-resident candidates
        unsigned long long best = 0ull;
#pragma unroll
        for (int k = 0; k < 16; ++k) {
            const int p = tid + (k << 10);
            const unsigned int fb = __float_as_uint(mind[k]);   // mind >= 0 -> monotonic bits
            const unsigned long long key =
                ((unsigned long long)fb << 32) | (unsigned long long)(~(unsigned int)p);
            best = (key > best) ? key : best;
        }
        // wave32 reduction
#pragma unroll
        for (int off = 16; off >= 1; off >>= 1) {
            const unsigned long long o = shfl_xor_u64(best, off);
            best = (o > best) ? o : best;
        }
        if ((tid & 31) == 0) red[tid >> 5] = best;
        __syncthreads();
        if (tid < 32) {
            unsigned long long v = red[tid];
#pragma unroll
            for (int off = 16; off >= 1; off >>= 1) {
                const unsigned long long o = shfl_xor_u64(v, off);
                v = (o > v) ? o : v;
            }
            if (tid == 0) {
                const int wi = (int)(~(unsigned int)(v & 0xffffffffull));
                winner  = wi;
                out[it] = wi;
            }
        }
        __syncthreads();

        const int   wi = winner;
        const float wx = pts[wi * 3 + 0];     // broadcast load, L2-resident
        const float wy = pts[wi * 3 + 1];
        const float wz = pts[wi * 3 + 2];
#pragma unroll
        for (int k = 0; k < 16; ++k) {
            const float dx = px[k] - wx, dy = py[k] - wy, dz = pz[k] - wz;
            const float d = dx * dx + dy * dy + dz * dz;
            mind[k] = fminf(mind[k], d);
        }
    }
}

// ---------------------------------------------------------------------------
// Kernel 2: gather sampled centroids (lc_xyz, lc_feat). One block per (b,g).
// ---------------------------------------------------------------------------
__global__ __launch_bounds__(64) void gather_centroids(const float* __restrict__ xyz,
                                                       const float* __restrict__ feat,
                                                       const int* __restrict__ fps_idx,
                                                       float* __restrict__ lc_xyz,
                                                       float* __restrict__ lc_feat) {
    const int bg  = blockIdx.x;                  // 0 .. B*G-1
    const int b   = bg >> 10;
    const int tid = threadIdx.x;                 // 0..63
    const int p   = fps_idx[bg];
    if (tid < 3)
        lc_xyz[(size_t)bg * 3 + tid] = xyz[((size_t)b * Nn + p) * 3 + tid];
    lc_feat[(size_t)bg * 64 + tid] = feat[((size_t)b * Nn + p) * 64 + tid];
}

// ---------------------------------------------------------------------------
// sorted top-32 insertion. Branchless compare-swap chain: every level is
// v_cmp + 4x v_cndmask (arrays stay in VGPRs, no divergent block copies).
// Outer guard is the only branch and skips the chain in the common case.
// ---------------------------------------------------------------------------
__device__ __forceinline__ void insert32(float (&bd)[Kn], int (&bi)[Kn], float d, int idx) {
    if (d < bd[Kn - 1]) {                      // rarely true after warmup
        float cd = d;
        int   ci = idx;
#pragma unroll
        for (int k = 0; k < Kn; ++k) {
            const bool  sw = cd < bd[k];
            const float tv = bd[k];
            const int   ti = bi[k];
            bd[k] = sw ? cd : tv;
            bi[k] = sw ? ci : ti;
            cd    = sw ? tv : cd;
            ci    = sw ? ti : ci;
        }
    }
}

// ---------------------------------------------------------------------------
// Kernel 3: kNN via WMMA distance tiles + register-resident top-32 selection.
// One 256-thread block per 16 centroids.
//
// Data movement: 1024-point superchunks (12 KB) are staged into LDS with
// GLOBAL_LOAD_ASYNC_TO_LDS_B128 (ASYNCcnt path), then 8 waves each compute a
// 16x16 d^2 tile per 128-point chunk with V_WMMA_F32_16X16X4_F32:
//   A[m, :] = (-2cx, -2cy, -2cz, 1)        (16x4)
//   B[:, n] = ( px ,  py ,  pz , |p|^2 )   (4x16)
//   C[m, n] = |c_m|^2
//   D = A*B + C = |c|^2 + |p|^2 - 2 c.p
// ---------------------------------------------------------------------------
__global__ __launch_bounds__(256) void knn_kernel(const float* __restrict__ xyz,
                                                  const float* __restrict__ feat,
                                                  const float* __restrict__ lc_xyz,
                                                  float* __restrict__ knn_xyz,
                                                  float* __restrict__ knn_feat) {
    const int b    = blockIdx.x >> 6;            // 64 tiles per batch
    const int tile = blockIdx.x & 63;
    const int tid  = threadIdx.x;                // 0..255
    const int lane = tid & 31;
    const int wave = tid >> 5;                   // 0..7

    const float* pts = xyz  + (size_t)b * Nn * 3;
    const float* fts = feat + (size_t)b * Nn * 64;

    __shared__ float s_cx[16], s_cy[16], s_cz[16], s_cn[16];
    __shared__ float s_pts[1024 * 3];            // 12 KB superchunk staging
    __shared__ float s_d2[16 * 128];             // 8 KB distance staging
    __shared__ int   s_knn[16 * Kn];

    if (tid < 16) {
        const int g = tile * 16 + tid;
        const float* c = lc_xyz + ((size_t)b * Gn + g) * 3;
        const float x = c[0], y = c[1], z = c[2];
        s_cx[tid] = x; s_cy[tid] = y; s_cz[tid] = z;
        s_cn[tid] = x * x + y * y + z * z;
    }
    __syncthreads();

    // A operand: 16x4 f32. lanes 0-15 -> (K=0,K=1); lanes 16-31 -> (K=2,K=3).
    v2f amat;
    {
        const int m = lane & 15;
        if (lane < 16) { amat[0] = -2.0f * s_cx[m]; amat[1] = -2.0f * s_cy[m]; }
        else           { amat[0] = -2.0f * s_cz[m]; amat[1] = 1.0f; }
    }
    // C init: VGPR j -> rows j (lanes 0-15) / j+8 (lanes 16-31), any column.
    const int rb = (lane >> 4) * 8;
    v8f cinit;
#pragma unroll
    for (int j = 0; j < 8; ++j) cinit[j] = s_cn[rb + j];

    // per-selector sorted ascending top-32 (registers via full unroll)
    float bd[Kn];
    int   bi[Kn];
#pragma unroll
    for (int k = 0; k < Kn; ++k) { bd[k] = 3.4e38f; bi[k] = 0; }

    const int row = tid;          // selector row when tid < 16
    const int col = lane & 15;

    for (int sc = 0; sc < Nn / 1024; ++sc) {     // 16 superchunks
        // ---- async stage 1024 points (3072 dwords) into LDS, 16B at a time
        const float* src = pts + (size_t)sc * 1024 * 3;
#pragma unroll
        for (int i = 0; i < 3; ++i) {
            const int e = tid + i * 256;         // 768 B128 transfers total
            const unsigned long long ga =
                (unsigned long long)(uintptr_t)(src + (size_t)e * 4);
            const unsigned la = (unsigned)(uintptr_t)(&s_pts[e * 4]);
            asm volatile("global_load_async_to_lds_b128 %0, %1, off"
                         :: "v"(la), "v"(ga) : "memory");
        }
        // warm the next superchunk into L2 while this one is consumed
        if (sc + 1 < Nn / 1024)
            __builtin_prefetch(src + 1024 * 3, 0, 0);
        asm volatile("s_wait_asynccnt 0" ::: "memory");
        __syncthreads();                          // all waves' copies visible

        for (int cc = 0; cc < 8; ++cc) {          // 8 chunks of 128 points
            // B operand from LDS (paired lane halves read same point)
            const int lp = cc * 128 + wave * 16 + col;
            const float x = s_pts[lp * 3 + 0];
            const float y = s_pts[lp * 3 + 1];
            const float z = s_pts[lp * 3 + 2];
            v2f bmat;
            if (lane < 16) { bmat[0] = x; bmat[1] = y; }
            else           { bmat[0] = z; bmat[1] = x * x + y * y + z * z; }

            v8f d = __builtin_amdgcn_wmma_f32_16x16x4_f32(
                /*neg_a=*/false, amat, /*neg_b=*/false, bmat,
                /*c_mod=*/(short)0, cinit, /*reuse_a=*/false, /*reuse_b=*/false);

            const int cg = wave * 16 + col;
#pragma unroll
            for (int j = 0; j < 8; ++j)
                s_d2[(rb + j) * 128 + cg] = d[j];

            __syncthreads();

            if (tid < 16) {
                const int base = sc * 1024 + cc * 128;
                const float4* rp = (const float4*)&s_d2[row * 128];
#pragma unroll 1
                for (int c4 = 0; c4 < 32; ++c4) { // ds_load_b128 per 4 cands
                    const float4 v = rp[c4];
                    const int ib = base + c4 * 4;
                    insert32(bd, bi, v.x, ib + 0);
                    insert32(bd, bi, v.y, ib + 1);
                    insert32(bd, bi, v.z, ib + 2);
                    insert32(bd, bi, v.w, ib + 3);
                }
            }
            __syncthreads();                      // also fences s_pts reuse
        }
    }

    if (tid < 16) {
#pragma unroll
        for (int k = 0; k < Kn; ++k) s_knn[row * Kn + k] = bi[k];
    }
    __syncthreads();

    // cooperative, channel-contiguous output gathers
    for (int e = tid; e < 16 * Kn * 3; e += 256) {
        const int r = e / (Kn * 3);
        const int k = (e / 3) % Kn;
        const int c = e % 3;
        const int p = s_knn[r * Kn + k];
        const int g = tile * 16 + r;
        knn_xyz[(((size_t)b * Gn + g) * Kn + k) * 3 + c] = pts[(size_t)p * 3 + c];
    }
    for (int e = tid; e < 16 * Kn * 64; e += 256) {
        const int r = e >> 11;
        const int k = (e >> 6) & (Kn - 1);
        const int c = e & 63;
        const int p = s_knn[r * Kn + k];
        const int g = tile * 16 + r;
        knn_feat[(((size_t)b * Gn + g) * Kn + k) * 64 + c] = fts[(size_t)p * 64 + c];
    }
}

// ---------------------------------------------------------------------------
// Host launcher
// ---------------------------------------------------------------------------
extern "C" void kernel_launch(void* const* d_in, const int* in_sizes, int n_in,
                              void* d_out, int out_size, void* d_ws, size_t ws_size,
                              hipStream_t stream) {
    const float* xyz  = (const float*)d_in[0];   // [4,16384,3]
    const float* feat = (const float*)d_in[1];   // [4,16384,64]

    float* out = (float*)d_out;
    float* lc_xyz   = out;                                         // [4,1024,3]
    float* lc_feat  = out + (size_t)Bn * Gn * 3;                   // [4,1024,64]
    float* knn_xyz  = lc_feat + (size_t)Bn * Gn * 64;              // [4,1024,32,3]
    float* knn_feat = knn_xyz + (size_t)Bn * Gn * Kn * 3;          // [4,1024,32,64]

    int* fps_idx = (int*)d_ws;                                     // [4,1024]

    fps_kernel<<<Bn, 1024, 0, stream>>>(xyz, fps_idx);
    gather_centroids<<<Bn * Gn, 64, 0, stream>>>(xyz, feat, fps_idx, lc_xyz, lc_feat);
    knn_kernel<<<(Bn * Gn) / 16, 256, 0, stream>>>(xyz, feat, lc_xyz, knn_xyz, knn_feat);
}